// TemporalDeConvBlock_80367428043219
// MI455X (gfx1250) — compile-verified
//
#include <hip/hip_runtime.h>
#include <stdint.h>

// Problem constants (match reference)
#define B_  32
#define T_  4096
#define C_  128
#define F_  128
#define K_  3
#define D_  8

typedef __attribute__((ext_vector_type(2))) float v2f;
typedef __attribute__((ext_vector_type(8))) float v8f;
typedef __attribute__((ext_vector_type(4))) unsigned int v4u;
typedef __attribute__((ext_vector_type(8))) unsigned int v8u;

// LDS weight layout: 384 rows (c*K+k) of 128 floats, padded to a 132-float
// pitch. The pitch is produced directly by the TDM: pad_interval=6 (pad after
// every 128 DWORDs) + pad_amount=3 (insert 4 DWORDs) => 528B = 132-float rows.
// Row deltas mod 64 banks: +1 row = 132 % 64 = 4, +3 rows (next channel) =
// 396 % 64 = 12, +6 rows = 24 -> the two 16-lane halves of each ds_load_b32
// touch disjoint bank ranges. Conflict-free.
#define LDS_ROW    132
#define LDS_FLOATS ((C_ * K_) * LDS_ROW)   // 384*132 = 50688 floats = 202752 B

template <bool RESIDUAL>
__global__ __launch_bounds__(256, 1)
void deconv_wmma_f32(const float* __restrict__ xin,   // (B,T,C) layer input
                     const float* __restrict__ wgt,   // (C,K,F)
                     const float* __restrict__ bias,  // (F,)
                     const float* __restrict__ xres,  // residual (B,T,F) or unused
                     float* __restrict__ out)         // (B,T,F)
{
    __shared__ float wlds[LDS_FLOATS];

    // ---- Stage all weights (C*K x F fp32 = 192 KB) into LDS with the TDM ----
    // 2D tensor: dim0 = 128 elements (one F row), dim1 = 384 rows (c*K+k),
    // 4-byte elements, row stride 128, LDS padding 4 DWORDs per 128 DWORDs.
    if (threadIdx.x < 32) {            // one wave issues the DMA (EXEC ignored)
        const unsigned long long ga = (unsigned long long)(uintptr_t)wgt;
        const unsigned int lds_base =
            (unsigned int)(uintptr_t)(void*)&wlds[0]; // LDS aperture: addr[31:0]

        v4u g0;
        g0[0] = 1u;                                  // count=1, user descriptor
        g0[1] = lds_base;                            // lds_addr (bytes)
        g0[2] = (unsigned int)(ga & 0xFFFFFFFFu);    // global_addr[31:0]
        g0[3] = (unsigned int)((ga >> 32) & 0x01FFFFFFu)  // global_addr[56:32]
              | (2u << 30);                          // type = 2 ("image")

        v8u g1;
        g1[0] = (2u << 16)      // data_size = 4 bytes
              | (1u << 20)      // pad_enable
              | (6u << 22)      // pad_interval: 128 DWORDs
              | (3u << 25);     // pad_amount:   4 DWORDs
        g1[1] = (128u << 16);   // tensor_dim0 = 128 (bits 79:48, low half)
        g1[2] = (384u << 16);   // tensor_dim1 = 384 (bits 111:80, low half)
        g1[3] = (128u << 16);   // tile_dim0   = 128 (bits 127:112)
        g1[4] = 384u;           // tile_dim1   = 384 (bits 143:128)
        g1[5] = 128u;           // tensor_dim0_stride = 128 (bits 207:160)
        g1[6] = 0u;
        g1[7] = 0u;

        // 2-operand form: VADDR2/VADDR3 = NULL (tensor <= 2D)
        asm volatile("tensor_load_to_lds %0, %1" :: "s"(g0), "s"(g1) : "memory");
        __builtin_amdgcn_s_wait_tensorcnt(0);
    }
    __syncthreads();

    const int lane = threadIdx.x & 31;
    const int wave = threadIdx.x >> 5;
    const int strip = blockIdx.x * 8 + wave;          // 8 waves/block, 8192 strips
    const int tiles_per_b = T_ / 16;                  // 256
    const int b      = strip >> 8;                    // strip / 256
    const int t_base = (strip & (tiles_per_b - 1)) * 16;

    const int g = lane >> 4;   // lane half: K-col pair {c0,c0+1} vs {c0+2,c0+3}
    const int l = lane & 15;   // row (A) / column (B,C,D) within 16-wide tile

    v8f acc[8];
#pragma unroll
    for (int f = 0; f < 8; ++f) acc[f] = (v8f)(0.0f);

#pragma unroll
    for (int k = 0; k < K_; ++k) {
        // matrix row r=l maps to time index t = t_base + l + (K-1)*D - k*D
        const int trow = t_base + l + (K_ - 1) * D_ - k * D_;
        const bool ok = (trow < T_);                  // rows >= T contribute 0
        const float* xp = xin + ((size_t)b * T_ + (size_t)trow) * C_ + 2 * g;
        const float* wp = &wlds[(2 * g * K_ + k) * LDS_ROW + l];

        for (int c0 = 0; c0 < C_; c0 += 4) {
            v2f a;
            if (ok) {
                a = *(const v2f*)(xp + c0);           // global_load_b64
            } else {
                a.x = 0.0f; a.y = 0.0f;
            }
#pragma unroll
            for (int f = 0; f < 8; ++f) {
                v2f bm;
                bm.x = wp[f * 16];                    // W row cc   (ds_load_b32)
                bm.y = wp[K_ * LDS_ROW + f * 16];     // W row cc+1 (ds_load_b32)
                acc[f] = __builtin_amdgcn_wmma_f32_16x16x4_f32(
                    /*neg_a=*/false, a, /*neg_b=*/false, bm,
                    /*c_mod=*/(short)0, acc[f],
                    /*reuse_a=*/false, /*reuse_b=*/false);
            }
            wp += 4 * K_ * LDS_ROW;                   // advance 4 channels
        }
    }

    // Epilogue: D layout — VGPR r holds rows r (lanes 0-15) / r+8 (lanes 16-31),
    // column l within the F-tile. Add nvalid*C*bias, ReLU, optional residual+ReLU.
#pragma unroll
    for (int ft = 0; ft < 8; ++ft) {
        const int fcol = ft * 16 + l;
        const float bv = bias[fcol] * (float)C_;
#pragma unroll
        for (int r = 0; r < 8; ++r) {
            const int row = t_base + r + 8 * g;
            const int nv = 1 + (row < T_ - D_) + (row < T_ - 2 * D_);
            float v = acc[ft][r] + (float)nv * bv;
            v = fmaxf(v, 0.0f);
            const size_t idx = ((size_t)b * T_ + (size_t)row) * F_ + fcol;
            if constexpr (RESIDUAL) {
                v = fmaxf(v + xres[idx], 0.0f);
            }
            out[idx] = v;
        }
    }
}

extern "C" void kernel_launch(void* const* d_in, const int* in_sizes, int n_in,
                              void* d_out, int out_size, void* d_ws, size_t ws_size,
                              hipStream_t stream) {
    const float* x  = (const float*)d_in[0];   // (B,T,C)
    const float* w1 = (const float*)d_in[1];   // (C,K,F)
    const float* b1 = (const float*)d_in[2];   // (F,)
    const float* w2 = (const float*)d_in[3];   // (F,K,F)
    const float* b2 = (const float*)d_in[4];   // (F,)
    float* out = (float*)d_out;                // (B,T,F)
    float* h1  = (float*)d_ws;                 // needs B*T*F*4 = 64 MB scratch

    const int strips = B_ * (T_ / 16);         // 8192
    dim3 grid(strips / 8);                     // 8 waves (strips) per block
    dim3 block(256);

    // layer 1: h1 = relu(deconv(x, W1, b1))
    deconv_wmma_f32<false><<<grid, block, 0, stream>>>(x, w1, b1, nullptr, h1);
    // layer 2 fused: out = relu(relu(deconv(h1, W2, b2)) + x)
    deconv_wmma_f32<true ><<<grid, block, 0, stream>>>(h1, w2, b2, x, out);
}